// RGCNLayer_26568667693632
// MI455X (gfx1250) — compile-verified
//
#include <hip/hip_runtime.h>
#include <hip/hip_bf16.h>

// Problem constants (match reference)
#define NN   20000   // nodes
#define EE   200000  // edges
#define INF  128     // in features
#define OUTF 128     // out features
#define NB   8       // bases
#define SI   16
#define SO   16

typedef __attribute__((ext_vector_type(2))) float v2f;
typedef __attribute__((ext_vector_type(8))) float v8f;

// ---------------------------------------------------------------------------
// Kernel 0: zero the aggregation buffer (d_ws)
// ---------------------------------------------------------------------------
__global__ __launch_bounds__(256) void zero_kernel(float* __restrict__ p, int n) {
    int i = blockIdx.x * 256 + threadIdx.x;
    if (i < n) p[i] = 0.0f;
}

// ---------------------------------------------------------------------------
// Kernel 1: per-edge message + atomic scatter-add into agg[N,128]
//   2 edges per 256-thread block, 128 threads per edge (one output feature
//   per thread). h[src] is staged in LDS so the 16 threads of a block-b only
//   touch LDS for the shared source subvector. W reads are coalesced
//   (consecutive f -> consecutive addresses) and hit L2 (total W = 1.9 MB).
// ---------------------------------------------------------------------------
__global__ __launch_bounds__(256) void edge_kernel(
        const float* __restrict__ h,
        const float* __restrict__ edge_norm,
        const float* __restrict__ weight,
        const int*   __restrict__ esrc,
        const int*   __restrict__ edst,
        const int*   __restrict__ etype,
        float*       __restrict__ agg) {
    const int sub = threadIdx.x >> 7;            // 0..1: edge slot in block
    const int f   = threadIdx.x & 127;           // output feature
    const int e   = blockIdx.x * 2 + sub;

    __shared__ float srcfeat[2][INF];
    {
        const int s = esrc[e];
        srcfeat[sub][f] = h[(size_t)s * INF + f];
    }
    __syncthreads();

    const int   t = etype[e];
    const int   b = f >> 4;                      // block index
    const int   o = f & 15;                      // out index within block
    const float* __restrict__ W  = weight + (size_t)t * (NB * SI * SO) + b * (SI * SO) + o;
    const float* __restrict__ sv = &srcfeat[sub][b * SI];

    __builtin_prefetch(W, 0, 3);                 // global_prefetch_b8: pull W row set

    float acc = 0.0f;
#pragma unroll
    for (int i = 0; i < SI; ++i)
        acc = __builtin_fmaf(sv[i], W[i * SO], acc);

    acc *= edge_norm[e];
    atomicAdd(&agg[(size_t)edst[e] * OUTF + f], acc);   // global_atomic_add_f32
}

// ---------------------------------------------------------------------------
// Kernel 2: loop_message = h @ loop_weight via V_WMMA_F32_16X16X4_F32,
// fused epilogue: out = relu(agg*node_norm + bias + loop_message).
//   Block = 256 threads = 8 waves; block b owns node rows [16b,16b+16);
//   wave w owns output-column tile [16w,16w+16). K=128 fully unrolled ->
//   32 WMMA issues per wave. A tile staged in LDS (stride 132 to avoid
//   64-bank conflicts at stride 128).
//
// Fragment layouts per CDNA5 ISA 7.12.2 (fp32, wave32):
//   A 16x4 : lanes 0-15 row=lane, v0=K0,v1=K1 ; lanes 16-31 v0=K2,v1=K3
//   B 4x16 : lanes 0-15 col=lane, v0=K0,v1=K1 ; lanes 16-31 v0=K2,v1=K3
//   C 16x16: VGPR j -> row j (lanes 0-15) / row j+8 (lanes 16-31), col=lane&15
// ---------------------------------------------------------------------------
#define LDSTRIDE 132
__global__ __launch_bounds__(256) void loop_fuse_kernel(
        const float* __restrict__ h,
        const float* __restrict__ lw,        // loop_weight [128,128] row-major
        const float* __restrict__ bias,      // [128]
        const float* __restrict__ agg,       // [N,128]
        const float* __restrict__ node_norm, // [N]
        float*       __restrict__ out) {     // [N,128]
    const int lane = threadIdx.x & 31;
    const int wave = threadIdx.x >> 5;       // 0..7 -> column tile
    const int row0 = blockIdx.x * 16;
    const int col0 = wave * 16;
    const int half = lane >> 4;              // 0: K lo pair, 1: K hi pair
    const int l16  = lane & 15;

    __shared__ float Atile[16 * LDSTRIDE];
    for (int i = threadIdx.x; i < 16 * INF; i += 256) {
        const int r = i >> 7, c = i & 127;
        Atile[r * LDSTRIDE + c] = h[(size_t)(row0 + r) * INF + c];
    }
    __syncthreads();

    v8f c = {};
#pragma unroll
    for (int k0 = 0; k0 < INF; k0 += 4) {
        const int ka = k0 + half * 2;
        v2f a, b;
        a.x = Atile[l16 * LDSTRIDE + ka];
        a.y = Atile[l16 * LDSTRIDE + ka + 1];
        b.x = lw[(size_t)ka       * OUTF + col0 + l16];
        b.y = lw[(size_t)(ka + 1) * OUTF + col0 + l16];
        // 8 args: (neg_a, A, neg_b, B, c_mod, C, reuse_a, reuse_b)
        c = __builtin_amdgcn_wmma_f32_16x16x4_f32(false, a, false, b,
                                                  (short)0, c, false, false);
    }

#pragma unroll
    for (int j = 0; j < 8; ++j) {
        const int r   = row0 + j + half * 8;
        const int col = col0 + l16;
        float v = c[j] + bias[col] + agg[(size_t)r * OUTF + col] * node_norm[r];
        out[(size_t)r * OUTF + col] = v > 0.0f ? v : 0.0f;
    }
}

// ---------------------------------------------------------------------------
// Kernel 3: time embedding gather: out2[n, :] = time_embed[time_idx[n], :]
// ---------------------------------------------------------------------------
__global__ __launch_bounds__(128) void time_kernel(
        const float* __restrict__ te,
        const int*   __restrict__ tidx,
        float*       __restrict__ out2) {
    const int n = blockIdx.x;
    const int f = threadIdx.x;
    out2[(size_t)n * INF + f] = te[(size_t)tidx[n] * INF + f];
}

// ---------------------------------------------------------------------------
extern "C" void kernel_launch(void* const* d_in, const int* in_sizes, int n_in,
                              void* d_out, int out_size, void* d_ws, size_t ws_size,
                              hipStream_t stream) {
    (void)in_sizes; (void)n_in; (void)out_size; (void)ws_size;

    const float* h          = (const float*)d_in[0];
    const float* edge_norm  = (const float*)d_in[1];
    const float* node_norm  = (const float*)d_in[2];
    const float* weight     = (const float*)d_in[3];
    const float* h_bias     = (const float*)d_in[4];
    const float* loop_weight= (const float*)d_in[5];
    const float* time_embed = (const float*)d_in[6];
    const int*   edge_src   = (const int*)d_in[7];
    const int*   edge_dst   = (const int*)d_in[8];
    const int*   edge_type  = (const int*)d_in[9];
    const int*   time_idx   = (const int*)d_in[10];

    float* out  = (float*)d_out;                  // node_repr [N,128] then time_emb [N,128]
    float* agg  = (float*)d_ws;                   // N*128 floats of scratch

    zero_kernel<<<(NN * OUTF + 255) / 256, 256, 0, stream>>>(agg, NN * OUTF);
    edge_kernel<<<EE / 2, 256, 0, stream>>>(h, edge_norm, weight,
                                            edge_src, edge_dst, edge_type, agg);
    loop_fuse_kernel<<<NN / 16, 256, 0, stream>>>(h, loop_weight, h_bias,
                                                  agg, node_norm, out);
    time_kernel<<<NN, 128, 0, stream>>>(time_embed, time_idx, out + (size_t)NN * OUTF);
}